// gumbel_wrapper_32126355374691
// MI455X (gfx1250) — compile-verified
//
#include <hip/hip_runtime.h>
#include <hip/hip_bf16.h>
#include <math.h>

// ---------------------------------------------------------------------------
// Gumbel top-K autoregressive sampler (K=2 beam, N=8192, L=64, D=128)
// CDNA5 / gfx1250. Core GEMM uses V_WMMA_F32_16X16X4_F32 (f32 kept to match
// reference numerics). W_h is pre-transposed once so both A and B fragments
// load as contiguous b64 per lane; each wave computes a 16x32 tile (two
// accumulators) to reuse A across column tiles. Sort is a 16384-key bitonic
// sort fully in LDS (128KB of the WGP's 320KB).
// ---------------------------------------------------------------------------

#define NN   8192
#define KK   2
#define LL   64
#define DD   128
#define MSORT (NN * KK)       // 16384 keys per step

typedef __attribute__((ext_vector_type(2))) float v2f;
typedef __attribute__((ext_vector_type(8))) float v8f;

// ------------------------------- RNG ---------------------------------------
__device__ __forceinline__ unsigned int avalanche(unsigned int x) {
  x ^= x >> 16; x *= 0x7FEB352Du;
  x ^= x >> 15; x *= 0x846CA68Bu;
  x ^= x >> 16;
  return x;
}
__device__ __forceinline__ float gumbel_noise(unsigned int seed, int step, int id) {
  unsigned int x = avalanche(seed ^ 0x9E3779B9u);
  x = avalanche(x ^ (unsigned int)(step * 0x85EBCA6B));
  x = avalanche(x ^ (unsigned int)(id   * 0xC2B2AE35));
  // uniform in (0,1), then gumbel
  float u = ((x >> 8) + 0.5f) * (1.0f / 16777216.0f);
  return -__logf(-__logf(u));
}

// --------------------------- E = W_emb @ W_x --------------------------------
__global__ __launch_bounds__(256) void compute_E(const float* __restrict__ W_emb,
                                                 const float* __restrict__ W_x,
                                                 float* __restrict__ E) {
  int i = blockIdx.x * blockDim.x + threadIdx.x;   // 0..255
  if (i >= KK * DD) return;
  int k = i >> 7, d = i & (DD - 1);
  float s = 0.f;
  for (int j = 0; j < DD; ++j) s += W_emb[k * DD + j] * W_x[j * DD + d];
  E[i] = s;
}

// ----------------- WhT[c][k] = Wh[k][c]  (once per launch) ------------------
__global__ __launch_bounds__(256) void transpose_Wh(const float* __restrict__ Wh,
                                                    float* __restrict__ WhT) {
  int i = blockIdx.x * blockDim.x + threadIdx.x;   // 0..16383
  int k = i >> 7, c = i & (DD - 1);
  WhT[(size_t)c * DD + k] = Wh[(size_t)k * DD + c];
}

// ------------------------------ init state ----------------------------------
__global__ void init_state(float* __restrict__ h_cur, int* __restrict__ toks0,
                           float* __restrict__ lg_sorted, float* __restrict__ g_sorted) {
  int i = blockIdx.x * blockDim.x + threadIdx.x;
  if (i < NN * DD) h_cur[i] = 0.f;
  if (i < NN) toks0[i] = 0;
  if (i < NN * KK) {
    float v = (i == 0) ? 0.f : -__builtin_inff();
    lg_sorted[i] = v;
    g_sorted[i]  = v;
  }
}

// ---------------- RNN step: h_new = tanh(E[tok] + h@W_h + b) -----------------
// One wave per 16x32 output tile; K=128 via 2x32 V_WMMA_F32_16X16X4_F32.
// A (16x4 f32): lane l holds A[m=l%16][kk + 2*(l/16) + {0,1}] in {a.x,a.y}
// B (4x16 f32): lane l holds B[kk + 2*(l/16) + {0,1}][n=l%16] -> contiguous
//               b64 from WhT[n][k..k+1] (mirror of A's K arrangement)
// C/D (16x16 f32, 8 VGPRs): acc[v] = C[v + 8*(l/16)][l%16]
__global__ __launch_bounds__(32) void rnn_gemm_tanh(
    const float* __restrict__ h,   const float* __restrict__ WhT,
    const float* __restrict__ E,   const float* __restrict__ bvec,
    const int*   __restrict__ tok, float* __restrict__ h_new) {
  const int lane = threadIdx.x & 31;
  const int colTile = blockIdx.x & 3;        // 128/32 double-column tiles
  const int rowTile = blockIdx.x >> 2;       // NN/16 row tiles
  const int m  = lane & 15;
  const int hi = lane >> 4;                  // 0 or 1
  const int rowBase = rowTile * 16;
  const int colBase = colTile * 32;

  v8f acc0 = {};
  v8f acc1 = {};
  const float* aRow  = h   + (size_t)(rowBase + m) * DD;
  const float* bCol0 = WhT + (size_t)(colBase + m) * DD;
  const float* bCol1 = WhT + (size_t)(colBase + 16 + m) * DD;
  for (int kk = 0; kk < DD; kk += 4) {
    const int ka = kk + 2 * hi;
    v2f a  = *(const v2f*)(aRow  + ka);      // b64, reused for both tiles
    v2f b0 = *(const v2f*)(bCol0 + ka);      // b64, contiguous in WhT
    v2f b1 = *(const v2f*)(bCol1 + ka);
    acc0 = __builtin_amdgcn_wmma_f32_16x16x4_f32(
        false, a, false, b0, (short)0, acc0, false, false);
    acc1 = __builtin_amdgcn_wmma_f32_16x16x4_f32(
        false, a, false, b1, (short)0, acc1, false, false);
  }

  #pragma unroll
  for (int v = 0; v < 8; ++v) {
    const int row = rowBase + v + 8 * hi;
    const int etok = tok[row] * DD;
    const int c0 = colBase + m;
    const int c1 = colBase + 16 + m;
    float x0 = acc0[v] + E[etok + c0] + bvec[c0];
    float x1 = acc1[v] + E[etok + c1] + bvec[c1];
    h_new[(size_t)row * DD + c0] = tanhf(x0);
    h_new[(size_t)row * DD + c1] = tanhf(x1);
  }
}

// -------- logits -> log_softmax -> gumbel propagation (per sample) ----------
__global__ __launch_bounds__(256) void rnn_logits_gumbel(
    const float* __restrict__ h_new, const float* __restrict__ Wo,
    const float* __restrict__ bo,
    const float* __restrict__ lg_sorted, const float* __restrict__ g_sorted,
    float* __restrict__ lg_flat, float* __restrict__ g_flat,
    const int* __restrict__ seed_p, int step) {
  const int n = blockIdx.x * 8 + (threadIdx.x >> 5);
  const int lane = threadIdx.x & 31;
  float s0 = 0.f, s1 = 0.f;
  for (int d = lane; d < DD; d += 32) {
    float hv = h_new[(size_t)n * DD + d];
    s0 += hv * Wo[d * 2 + 0];
    s1 += hv * Wo[d * 2 + 1];
  }
  for (int off = 16; off; off >>= 1) {
    s0 += __shfl_down(s0, off, 32);
    s1 += __shfl_down(s1, off, 32);
  }
  if (lane == 0) {
    float l0 = s0 + bo[0], l1 = s1 + bo[1];
    float mx  = fmaxf(l0, l1);
    float lse = mx + __logf(__expf(l0 - mx) + __expf(l1 - mx));
    float lgprev = lg_sorted[n * 2];        // logits[:, :1]
    float gprev  = g_sorted[n * 2];         // gumbel[:, :1]
    float ln0 = lgprev + (l0 - lse);
    float ln1 = lgprev + (l1 - lse);
    unsigned int seed = (unsigned int)seed_p[0];
    float gn0 = ln0 + gumbel_noise(seed, step, n * 2 + 0);
    float gn1 = ln1 + gumbel_noise(seed, step, n * 2 + 1);
    float Z = fmaxf(gn0, gn1);              // nanmax-like for our domain
    float base = __expf(-gprev) - __expf(-Z);
    float gf0 = -__logf(base + __expf(-gn0));
    float gf1 = -__logf(base + __expf(-gn1));
    if (isnan(gf0)) gf0 = -__builtin_inff();
    if (isnan(gf1)) gf1 = -__builtin_inff();
    lg_flat[n * 2 + 0] = ln0;  lg_flat[n * 2 + 1] = ln1;
    g_flat [n * 2 + 0] = gf0;  g_flat [n * 2 + 1] = gf1;
  }
}

// --------- global descending argsort of 16384 gumbel keys (bitonic) ---------
// Stable-ish: ties broken by original index (smaller index wins), matching
// jnp.argsort's stability. Keys live entirely in LDS (128KB dynamic shared).
__global__ __launch_bounds__(1024) void sort_gumbel(const float* __restrict__ g_flat,
                                                    int* __restrict__ sortedIdx) {
  extern __shared__ unsigned long long sm[];
  const int tid = threadIdx.x;
  for (int i = tid; i < MSORT; i += 1024) {
    unsigned int bits = __float_as_uint(g_flat[i]);
    bits = (bits & 0x80000000u) ? ~bits : (bits | 0x80000000u);  // float->sortable
    sm[i] = ((unsigned long long)bits << 32) | (unsigned int)(~i);
  }
  __syncthreads();
  for (int size = 2; size <= MSORT; size <<= 1) {
    for (int stride = size >> 1; stride > 0; stride >>= 1) {
      for (int i = tid; i < MSORT / 2; i += 1024) {
        int lo  = ((i & ~(stride - 1)) << 1) | (i & (stride - 1));
        int hiI = lo + stride;
        bool desc = ((lo & size) == 0);      // overall descending sort
        unsigned long long a = sm[lo], b = sm[hiI];
        if (desc ? (a < b) : (a > b)) { sm[lo] = b; sm[hiI] = a; }
      }
      __syncthreads();
    }
  }
  for (int p = tid; p < MSORT; p += 1024)
    sortedIdx[p] = (int)(~(unsigned int)sm[p]);
}

// ---- reorder: sorted lg/g + beam-search parent/token chain for this step ---
__global__ __launch_bounds__(256) void reorder_step(
    const int* __restrict__ sortedIdx,
    const float* __restrict__ lg_flat, const float* __restrict__ g_flat,
    float* __restrict__ lg_sorted, float* __restrict__ g_sorted,
    int* __restrict__ parents_t, int* __restrict__ toks_next) {
  int p = blockIdx.x * blockDim.x + threadIdx.x;
  if (p >= MSORT) return;
  int k = p / NN, n = p - k * NN;
  int j = sortedIdx[p];
  lg_sorted[n * 2 + k] = lg_flat[j];
  g_sorted [n * 2 + k] = g_flat [j];
  if (k == 0) {
    parents_t[n] = j >> 1;     // (idx // K)[:N]
    toks_next[n] = j & 1;      // token chosen at this position
  }
}

// ------------------- h_cur[n] = h_new[parents_t[n]] -------------------------
__global__ __launch_bounds__(32) void gather_h(const float* __restrict__ h_new,
                                               const int* __restrict__ parents_t,
                                               float* __restrict__ h_cur) {
  const int n = blockIdx.x;
  const float4* src = (const float4*)(h_new + (size_t)parents_t[n] * DD);
  float4*       dst = (float4*)(h_cur + (size_t)n * DD);
  dst[threadIdx.x] = src[threadIdx.x];
}

// ------------------- backtrack the beam to emit s[:,0,:] --------------------
__global__ __launch_bounds__(256) void backtrack(const int* __restrict__ parents,
                                                 const int* __restrict__ toks,
                                                 float* __restrict__ out_s) {
  int n = blockIdx.x * blockDim.x + threadIdx.x;
  if (n >= NN) return;
  int cur = n;
  for (int t = LL - 1; t >= 0; --t) {
    out_s[(size_t)n * LL + t] = (float)toks[(size_t)(t + 1) * NN + cur];
    cur = parents[(size_t)t * NN + cur];
  }
}

// ---------------- lp, normalized weights, kappa (single block) --------------
__global__ __launch_bounds__(1024) void finalize(const float* __restrict__ lg_sorted,
                                                 const float* __restrict__ g_sorted,
                                                 float* __restrict__ out_lp,
                                                 float* __restrict__ out_w,
                                                 float* __restrict__ out_kappa,
                                                 float* __restrict__ w_tmp) {
  __shared__ float red[32];
  const int tid = threadIdx.x;
  const float kappa = g_sorted[1];           // g[0,1]
  float local = 0.f;
  for (int n = tid; n < NN; n += 1024) {
    float lp0 = lg_sorted[n * 2];
    out_lp[n] = lp0 * 0.5f;                  // LOGPROB_FACTOR
    float w = __expf(lp0) / (-expm1f(-__expf(lp0 - kappa)));
    if (isnan(w)) w = 0.f;
    w_tmp[n] = w;
    local += w;
  }
  for (int off = 16; off; off >>= 1) local += __shfl_down(local, off, 32);
  if ((tid & 31) == 0) red[tid >> 5] = local;
  __syncthreads();
  if (tid < 32) {
    float v = red[tid];
    for (int off = 16; off; off >>= 1) v += __shfl_down(v, off, 32);
    if (tid == 0) red[0] = v;
  }
  __syncthreads();
  const float total = red[0];
  for (int n = tid; n < NN; n += 1024) out_w[n] = w_tmp[n] / total;
  if (tid == 0) out_kappa[0] = kappa;
}

// ---------------------------------------------------------------------------
extern "C" void kernel_launch(void* const* d_in, const int* in_sizes, int n_in,
                              void* d_out, int out_size, void* d_ws, size_t ws_size,
                              hipStream_t stream) {
  const float* W_emb = (const float*)d_in[0];   // (2,128)
  const float* W_x   = (const float*)d_in[1];   // (128,128)
  const float* W_h   = (const float*)d_in[2];   // (128,128)
  const float* b     = (const float*)d_in[3];   // (128,)
  const float* W_o   = (const float*)d_in[4];   // (128,2)
  const float* b_o   = (const float*)d_in[5];   // (2,)
  const int*   seed  = (const int*)d_in[7];     // scalar

  // ---- workspace layout (all 256B aligned) ----
  char* ws = (char*)d_ws;
  size_t off = 0;
  auto alloc = [&](size_t bytes) { void* p = ws + off; off = (off + bytes + 255) & ~(size_t)255; return p; };
  float* h_cur     = (float*)alloc(sizeof(float) * NN * DD);
  float* h_new     = (float*)alloc(sizeof(float) * NN * DD);
  float* E         = (float*)alloc(sizeof(float) * KK * DD);
  float* WhT       = (float*)alloc(sizeof(float) * DD * DD);
  float* lg_sorted = (float*)alloc(sizeof(float) * NN * KK);
  float* g_sorted  = (float*)alloc(sizeof(float) * NN * KK);
  float* lg_flat   = (float*)alloc(sizeof(float) * MSORT);
  float* g_flat    = (float*)alloc(sizeof(float) * MSORT);
  int*   sortedIdx = (int*)  alloc(sizeof(int)   * MSORT);
  int*   parents   = (int*)  alloc(sizeof(int)   * LL * NN);        // row t
  int*   toks      = (int*)  alloc(sizeof(int)   * (LL + 1) * NN);  // row 0 = zeros
  float* w_tmp     = (float*)alloc(sizeof(float) * NN);
  (void)ws_size; (void)in_sizes; (void)n_in; (void)out_size;

  float* out_s     = (float*)d_out;            // N*L
  float* out_lp    = out_s + (size_t)NN * LL;  // N
  float* out_w     = out_lp + NN;              // N
  float* out_kappa = out_w + NN;               // 1

  compute_E<<<1, 256, 0, stream>>>(W_emb, W_x, E);
  transpose_Wh<<<(DD * DD) / 256, 256, 0, stream>>>(W_h, WhT);
  init_state<<<(NN * DD + 255) / 256, 256, 0, stream>>>(h_cur, toks, lg_sorted, g_sorted);

  for (int t = 0; t < LL; ++t) {
    rnn_gemm_tanh<<<(NN / 16) * (DD / 32), 32, 0, stream>>>(
        h_cur, WhT, E, b, toks + (size_t)t * NN, h_new);
    rnn_logits_gumbel<<<NN / 8, 256, 0, stream>>>(
        h_new, W_o, b_o, lg_sorted, g_sorted, lg_flat, g_flat, seed, t);
    sort_gumbel<<<1, 1024, MSORT * sizeof(unsigned long long), stream>>>(g_flat, sortedIdx);
    reorder_step<<<(MSORT + 255) / 256, 256, 0, stream>>>(
        sortedIdx, lg_flat, g_flat, lg_sorted, g_sorted,
        parents + (size_t)t * NN, toks + (size_t)(t + 1) * NN);
    gather_h<<<NN, 32, 0, stream>>>(h_new, parents + (size_t)t * NN, h_cur);
  }

  backtrack<<<NN / 256, 256, 0, stream>>>(parents, toks, out_s);
  finalize<<<1, 1024, 0, stream>>>(lg_sorted, g_sorted, out_lp, out_w, out_kappa, w_tmp);
}